// LIF_61263413510486
// MI455X (gfx1250) — compile-verified
//
#include <hip/hip_runtime.h>
#include <cstdint>

// LIF scan: o[t,n] via u = 0.5*(u + ir[t,n]); spike = (u >= 1); reset.
// Memory-bound streaming scan: 512 MiB traffic -> ~22.5 us floor @ 23.3 TB/s.
// Data path: async global->LDS (ASYNCcnt) double buffering, NT on both streams.

typedef __attribute__((ext_vector_type(4))) float f4;

#define T_STEPS 64
#define CHUNK   8
#define NCHUNK  (T_STEPS / CHUNK)
#define BLOCK   256

__device__ __forceinline__ void async_load_b128_nt(uint32_t lds_addr, uint32_t goff,
                                                   const float* base) {
    // GVS mode: saddr(64-bit SGPR pair) + vaddr(32-bit byte offset VGPR);
    // VDST VGPR holds the LDS byte address. Tracked by ASYNCcnt.
    // th:TH_LOAD_NT -> single-use stream, don't retain in L2 (192 MB < 256 MiB stream).
    asm volatile("global_load_async_to_lds_b128 %0, %1, %2 th:TH_LOAD_NT"
                 :: "v"(lds_addr), "v"(goff), "s"(base)
                 : "memory");
}

__global__ __launch_bounds__(BLOCK) void lif_async_kernel(
    const float* __restrict__ ir, float* __restrict__ out, unsigned n_per_t) {
    __shared__ f4 lds[2 * CHUNK * BLOCK];  // 64 KiB double buffer

    const unsigned tid     = blockIdx.x * BLOCK + threadIdx.x;
    const unsigned gbyte0  = tid * 16u;        // byte offset inside a timestep
    const unsigned tstride = n_per_t * 4u;     // bytes per timestep (4 MiB)

    // Low 32 bits of a generic pointer to __shared__ == LDS byte offset.
    const uint32_t lds_base       = (uint32_t)(uintptr_t)(&lds[threadIdx.x]);
    const uint32_t lds_buf_stride = CHUNK * BLOCK * (uint32_t)sizeof(f4);
    const uint32_t lds_t_stride   = BLOCK * (uint32_t)sizeof(f4);

    // Prologue: async-load chunk 0 into buffer 0.
#pragma unroll
    for (int k = 0; k < CHUNK; ++k)
        async_load_b128_nt(lds_base + (uint32_t)k * lds_t_stride,
                           gbyte0 + (uint32_t)k * tstride, ir);

    f4 u = {0.0f, 0.0f, 0.0f, 0.0f};

#pragma unroll
    for (int c = 0; c < NCHUNK; ++c) {
        if (c + 1 < NCHUNK) {
            // Prefetch chunk c+1 into the other buffer.
#pragma unroll
            for (int k = 0; k < CHUNK; ++k) {
                const unsigned t = (unsigned)(c + 1) * CHUNK + (unsigned)k;
                async_load_b128_nt(lds_base + (uint32_t)((c + 1) & 1) * lds_buf_stride
                                            + (uint32_t)k * lds_t_stride,
                                   gbyte0 + t * tstride, ir);
            }
            // 16 outstanding; async loads complete in order -> <=8 means chunk c landed.
            asm volatile("s_wait_asynccnt 8" ::: "memory");
        } else {
            asm volatile("s_wait_asynccnt 0" ::: "memory");
        }

        // Consume chunk c from LDS, run 8 LIF steps, stream results out (NT).
#pragma unroll
        for (int k = 0; k < CHUNK; ++k) {
            const int t = c * CHUNK + k;
            f4 x = lds[((c & 1) * CHUNK + k) * BLOCK + threadIdx.x];
            f4 o;
#pragma unroll
            for (int i = 0; i < 4; ++i) {
                float ut  = 0.5f * (u[i] + x[i]);   // du+u_prev = 0.5*(u_prev+ir)
                bool spike = (ut >= 1.0f);          // heaviside(u_t - U_TH)
                o[i] = spike ? 1.0f : 0.0f;
                u[i] = spike ? 0.0f : ut;           // reset to U_REST = 0
            }
            f4* outp = (f4*)(out + (size_t)t * n_per_t) + tid;
            __builtin_nontemporal_store(o, outp);
        }
    }
}

extern "C" void kernel_launch(void* const* d_in, const int* in_sizes, int n_in,
                              void* d_out, int out_size, void* d_ws, size_t ws_size,
                              hipStream_t stream) {
    const float* ir  = (const float*)d_in[0];
    float*       out = (float*)d_out;
    const unsigned total   = (unsigned)in_sizes[0];  // 64*32*32768 = 67,108,864
    const unsigned n_per_t = total / T_STEPS;        // 1,048,576 floats / timestep
    const unsigned threads = n_per_t / 4;            // 262,144 threads
    dim3 grid(threads / BLOCK);                      // 1024 blocks of 256 (8 waves)
    lif_async_kernel<<<grid, dim3(BLOCK), 0, stream>>>(ir, out, n_per_t);
}